// PluckettLuceKeibaBetting_77266461655153
// MI455X (gfx1250) — compile-verified
//
#include <hip/hip_runtime.h>
#include <cstdint>
#include <cstddef>

// Plackett-Luce keiba betting probabilities, MI455X (gfx1250, wave32).
// B=8192 batches x 18 horses -> 6360 outputs/batch (208 MB stores).
// Store-bandwidth bound (~9us floor @ 23.3 TB/s). Analytic factorization
// p(f,s,t) = W[f,s]*e_t removes all scatter atomics; outputs streamed as
// 16-byte global_store_b128.

#define NH        18
#define PAD       20     // padded LDS tile so WMMA A-loads need no guards
#define N_PAIR    153
#define N_OPAIR   306
#define N_TRIP    816
#define OUT_PER_B 6360   // 18+18+153+153+306+816+4896  (divisible by 4)
#define THREADS   256

typedef float v2f  __attribute__((ext_vector_type(2)));
typedef float v8f  __attribute__((ext_vector_type(8)));
typedef unsigned int u32x4 __attribute__((ext_vector_type(4)));
typedef int  i32x4 __attribute__((ext_vector_type(4)));
typedef int  i32x8 __attribute__((ext_vector_type(8)));

#if __has_builtin(__builtin_amdgcn_tensor_load_to_lds) && __has_builtin(__builtin_amdgcn_s_wait_tensorcnt)
#define HAVE_TDM 1
#pragma message("CDNA5 probe: tensor_load_to_lds builtin AVAILABLE")
#else
#pragma message("CDNA5 probe: tensor_load_to_lds builtin MISSING")
#endif

#if __has_builtin(__builtin_amdgcn_global_load_async_to_lds_b32) && __has_builtin(__builtin_amdgcn_s_wait_asynccnt)
#define HAVE_ASYNC 1
#pragma message("CDNA5 probe: global_load_async_to_lds_b32 builtin AVAILABLE")
#else
#pragma message("CDNA5 probe: global_load_async_to_lds_b32 builtin MISSING")
#endif

#if __has_builtin(__builtin_amdgcn_wmma_f32_16x16x4_f32)
#define HAVE_WMMA4 1
#pragma message("CDNA5 probe: wmma_f32_16x16x4_f32 builtin AVAILABLE")
#else
#pragma message("CDNA5 probe: wmma_f32_16x16x4_f32 builtin MISSING")
#endif

// ---- index decoders (itertools orderings) -------------------------------
__device__ __forceinline__ void pair_decode(int pid, int& a, int& b) {
  // combinations(range(18),2) lexicographic
  int aa = 0, cnt = 17;
  while (pid >= cnt) { pid -= cnt; ++aa; --cnt; }
  a = aa; b = aa + 1 + pid;
}
__device__ __forceinline__ void trip_decode(int t3, int& a, int& b, int& c) {
  // combinations(range(18),3) lexicographic
  int aa = 0;
  for (;;) { int cnt = ((17 - aa) * (16 - aa)) >> 1; if (t3 < cnt) break; t3 -= cnt; ++aa; }
  int bb = aa + 1;
  for (;;) { int cnt = 17 - bb; if (t3 < cnt) break; t3 -= cnt; ++bb; }
  a = aa; b = bb; c = bb + 1 + t3;
}

#ifdef HAVE_WMMA4
// Row-sums over k=0..19 (cols 18,19 are zero padding) of a PAD-strided 20x20
// LDS tile, for m=0..15, using V_WMMA_F32_16X16X4_F32 with B = all-ones
// (B layout irrelevant when every element is 1.0). A layout (ISA 7.12.2,
// 32-bit A 16x4): lanes 0-15 hold {K0,K1}, lanes 16-31 hold {K2,K3}.
// sel: 0 -> M[m][k] (row sums), 1 -> M[k][m] (col sums).
__device__ __forceinline__ void wmma_rowsum20(int lane,
                                              const float (*M)[PAD],
                                              int sel, float* dst) {
  v8f c = {0.f, 0.f, 0.f, 0.f, 0.f, 0.f, 0.f, 0.f};
  v2f bones = {1.0f, 1.0f};
  const int m    = lane & 15;
  const int koff = (lane >> 4) << 1;  // 0 for lanes 0-15, 2 for lanes 16-31
  for (int chunk = 0; chunk < 5; ++chunk) {
    int k0 = chunk * 4 + koff;        // <= 18, +1 <= 19: always in padded range
    v2f a;
    if (sel == 0) { a.x = M[m][k0];   a.y = M[m][k0 + 1]; }
    else          { a.x = M[k0][m];   a.y = M[k0 + 1][m]; }
    c = __builtin_amdgcn_wmma_f32_16x16x4_f32(false, a, false, bones,
                                              (short)0, c, false, false);
  }
  // C/D layout: VGPR i -> M=i (lanes 0-15), M=8+i (lanes 16-31); same all N.
  if (lane == 0)  { for (int i = 0; i < 8; ++i) dst[i]     = c[i]; }
  if (lane == 16) { for (int i = 0; i < 8; ++i) dst[8 + i] = c[i]; }
}
#endif

__global__ __launch_bounds__(THREADS)
void pl_keiba_kernel(const float* __restrict__ scores,
                     const int*   __restrict__ nrun,
                     float*       __restrict__ out, int B) {
  const int b   = blockIdx.x;
  const int tid = threadIdx.x;

  __shared__ float sRaw[NH];
  __shared__ float sE[NH];
  __shared__ float sAB[PAD][PAD];   // AB[f][s]; diag + pad rows/cols = 0
  __shared__ float sW[PAD][PAD];    // W[f][s] = AB/(D-ef-es)
  __shared__ float sRowW[NH], sColW[NH], sColAB[NH];
  __shared__ float sD, sInvD, sWtot;
  __shared__ int   sAddT;

  if (tid == 0) __builtin_prefetch(scores + (size_t)b * NH, 0, 0);

  // Zero-fill padded tiles (makes diag/padding contributions vanish).
  for (int k = tid; k < PAD * PAD; k += THREADS) {
    ((float*)sAB)[k] = 0.f;
    ((float*)sW)[k]  = 0.f;
  }

  // Authoritative per-lane load (async/TDM paths below write the same bytes).
  if (tid < NH) sRaw[tid] = scores[(size_t)b * NH + tid];

#ifdef HAVE_TDM
  if (tid < 32) {  // one wave issues one TDM op (EXEC-0 waves branch around)
    unsigned long long ga = (unsigned long long)(const void*)(scores + (size_t)b * NH);
    unsigned ldsOff = (unsigned)(unsigned long long)(const void*)&sRaw[0];  // low 32 = LDS offset
    // D# group 0: count=1 valid; lds_addr; global_addr[56:0]; type=2
    u32x4 g0 = { 1u, ldsOff,
                 (unsigned)(ga & 0xFFFFFFFFull),
                 ((unsigned)((ga >> 32) & 0x01FFFFFFull)) | 0x80000000u };
    // D# group 1: data_size=4B; tensor_dim0=18; tensor_dim1=1; tile=18x1;
    // tensor_dim0_stride=18; tensor_dim1_stride=18. (Bit offsets per ISA 8.4)
    i32x8 g1 = { 0x20000, (int)(18u << 16), (int)(1u << 16), (int)(18u << 16),
                 1, 18, (int)(18u << 16), 0 };
    i32x4 gz = { 0, 0, 0, 0 };
#if __clang_major__ >= 23
    i32x8 gz8 = { 0, 0, 0, 0, 0, 0, 0, 0 };
    __builtin_amdgcn_tensor_load_to_lds(g0, g1, gz, gz, gz8, 0);
#else
    __builtin_amdgcn_tensor_load_to_lds(g0, g1, gz, gz, 0);
#endif
    __builtin_amdgcn_s_wait_tensorcnt(0);
  }
#elif defined(HAVE_ASYNC)
  if (tid < NH) {
    __builtin_amdgcn_global_load_async_to_lds_b32(
        (const __attribute__((address_space(1))) void*)(scores + (size_t)b * NH + tid),
        (__attribute__((address_space(3))) void*)&sRaw[tid], 0, 0);
  }
  if (tid < 32) __builtin_amdgcn_s_wait_asynccnt(0);
#endif
  __syncthreads();

  // e_i = exp(clip(score, -10, 10))
  if (tid < NH) {
    float x = sRaw[tid];
    x = fminf(10.0f, fmaxf(-10.0f, x));
    sE[tid] = expf(x);
  }
  __syncthreads();

  if (tid == 0) {
    float d = 0.f;
    for (int i = 0; i < NH; ++i) d += sE[i];
    sD = d; sInvD = 1.0f / d;
    sAddT = (nrun[b] > 7) ? 1 : 0;   // fuku3 when more than 7 runners
  }
  __syncthreads();

  // AB[f][s] and W[f][s] over the 306 ordered pairs.
  for (int k = tid; k < N_OPAIR; k += THREADS) {
    int f = k / 17, m = k % 17;
    int s = (m < f) ? m : m + 1;
    float ef = sE[f], es = sE[s];
    float d2 = sD - ef;
    float ab = (ef * sInvD) * (es / d2);
    float d3 = d2 - es;
    sAB[f][s] = ab;
    sW[f][s]  = ab / d3;
  }
  __syncthreads();

  // rowW[f] = sum_s W[f][s], colW[s] = sum_f W[f][s], colAB[s] = sum_f AB[f][s]
#ifdef HAVE_WMMA4
  if (tid < 32) {                 // wave 0: WMMA for m = 0..15 of each reduction
    wmma_rowsum20(tid, sW, 0, sRowW);
    wmma_rowsum20(tid, sW, 1, sColW);
    wmma_rowsum20(tid, sAB, 1, sColAB);
  } else if (tid < 38) {          // scalar tail for indices 16, 17
    int t = tid - 32, i = 16 + (t & 1), which = t >> 1;
    float acc = 0.f;
    for (int k = 0; k < NH; ++k)
      acc += (which == 0) ? sW[i][k] : (which == 1) ? sW[k][i] : sAB[k][i];
    if (which == 0) sRowW[i] = acc;
    else if (which == 1) sColW[i] = acc;
    else sColAB[i] = acc;
  }
#else
  if (tid < 3 * NH) {
    int i = tid % NH, which = tid / NH;
    float acc = 0.f;
    for (int k = 0; k < NH; ++k)
      acc += (which == 0) ? sW[i][k] : (which == 1) ? sW[k][i] : sAB[k][i];
    if (which == 0) sRowW[i] = acc;
    else if (which == 1) sColW[i] = acc;
    else sColAB[i] = acc;
  }
#endif
  __syncthreads();
  if (tid == 0) {
    float wt = 0.f;
    for (int i = 0; i < NH; ++i) wt += sRowW[i];
    sWtot = wt;
  }
  __syncthreads();

  // Per-element closed form for output column j.
  auto outval = [&](int j) -> float {
    if (j < 18) {                                   // tansho: P(1st = i)
      return sE[j] * sInvD;
    } else if (j < 36) {                            // fukusho: top2 or top3
      int i = j - 18;
      float v = sE[i] * sInvD + sColAB[i];
      if (sAddT) v += sE[i] * (sWtot - sRowW[i] - sColW[i]);
      return v;
    } else if (j < 36 + N_PAIR) {                   // umaren
      int a, bb; pair_decode(j - 36, a, bb);
      return sAB[a][bb] + sAB[bb][a];
    } else if (j < 36 + 2 * N_PAIR) {               // wide: both in top 3
      int a, bb; pair_decode(j - (36 + N_PAIR), a, bb);
      return sAB[a][bb] + sAB[bb][a]
        + sE[bb] * (sRowW[a] - sW[a][bb]) + sE[a] * (sRowW[bb] - sW[bb][a])
        + sE[bb] * (sColW[a] - sW[bb][a]) + sE[a] * (sColW[bb] - sW[a][bb]);
    } else if (j < 342 + N_OPAIR) {                 // umatan
      int k = j - 342;
      int f = k / 17, m = k % 17;
      int s = (m < f) ? m : m + 1;
      return sAB[f][s];
    } else if (j < 648 + N_TRIP) {                  // sanrenpuku
      int a, bb, c; trip_decode(j - 648, a, bb, c);
      return sE[c]  * (sW[a][bb] + sW[bb][a])
           + sE[bb] * (sW[a][c]  + sW[c][a])
           + sE[a]  * (sW[bb][c] + sW[c][bb]);
    } else {                                        // sanrentan: W[f][s]*e_t
      int k4 = j - 1464;
      int k = k4 >> 4, r = k4 & 15;
      int f = k / 17, m = k % 17;
      int s = (m < f) ? m : m + 1;
      int lo = (f < s) ? f : s, hi = (f < s) ? s : f;
      int t = r;
      if (t >= lo) ++t;
      if (t >= hi) ++t;
      return sW[f][s] * sE[t];
    }
  };

  // Stream outputs as 16-byte vectors (row length 6360 = 4*1590, base 16B
  // aligned since 6360*4 % 16 == 0) -> global_store_b128, fully coalesced.
  float* o = out + (size_t)b * OUT_PER_B;
  for (int j4 = tid * 4; j4 < OUT_PER_B; j4 += THREADS * 4) {
    float4 v;
    v.x = outval(j4);
    v.y = outval(j4 + 1);
    v.z = outval(j4 + 2);
    v.w = outval(j4 + 3);
    *(float4*)(o + j4) = v;
  }
}

extern "C" void kernel_launch(void* const* d_in, const int* in_sizes, int n_in,
                              void* d_out, int out_size, void* d_ws, size_t ws_size,
                              hipStream_t stream) {
  const float* scores = (const float*)d_in[0];
  const int*   nrun   = (const int*)d_in[1];
  float*       out    = (float*)d_out;
  const int B = in_sizes[0] / NH;   // 8192
  pl_keiba_kernel<<<dim3(B), dim3(THREADS), 0, stream>>>(scores, nrun, out, B);
}